// RAA_63660005261554
// MI455X (gfx1250) — compile-verified
//
#include <hip/hip_runtime.h>
#include <hip/hip_bf16.h>
#include <math.h>

// Problem constants (B, R, N, E) = (4, 4, 256, 64)
#define BB 4
#define RR 4
#define NN 256
#define EE 64
#define LAMDA_C 0.5f
#define EPS_C 1e-6f

typedef __attribute__((ext_vector_type(2))) float v2f;
typedef __attribute__((ext_vector_type(8))) float v8f;

// ---------------------------------------------------------------------------
// Kernel 1: Q = n_emb @ Wq^T, K = n_emb @ Wk^T
// n_emb: (B*N, E) rows; Wq/Wk: (E, E) row-major. Tiny (8.4 MFLOP total).
// grid = B*N = 1024 blocks, block = 64 threads (one output column each).
// ---------------------------------------------------------------------------
__global__ __launch_bounds__(64) void qk_kernel(const float* __restrict__ n_emb,
                                                const float* __restrict__ Wq,
                                                const float* __restrict__ Wk,
                                                float* __restrict__ Q,
                                                float* __restrict__ K) {
    __shared__ float nrow[EE];
    const int i = blockIdx.x;      // 0 .. B*N-1
    const int e = threadIdx.x;     // 0 .. 63
    nrow[e] = n_emb[i * EE + e];
    __syncthreads();
    const float* wq = Wq + e * EE;
    const float* wk = Wk + e * EE;
    float aq = 0.0f, ak = 0.0f;
#pragma unroll 8
    for (int j = 0; j < EE; ++j) {
        const float nv = nrow[j];
        aq = fmaf(nv, wq[j], aq);
        ak = fmaf(nv, wk[j], ak);
    }
    Q[i * EE + e] = aq;
    K[i * EE + e] = ak;
}

// ---------------------------------------------------------------------------
// Kernel 2: ori[b,q,k] = (Q[b] @ K[b]^T)[q,k] / 4096  via V_WMMA_F32_16X16X4_F32
// One 16x16 output tile per wave; K-dim loop of 16 WMMA ops (E=64, K=4/op).
// A (16x4 f32): lanes 0-15 -> M=0..15, VGPR0=K0,VGPR1=K1; lanes 16-31 -> K2,K3.
// B (4x16): mirror layout; for Q@K^T the per-lane addressing into row-major K
// is identical to A's addressing into Q.
// D (16x16 f32): VGPR j -> M = 8*half + j, N = lane&15.
// grid = B*16*16/8 = 128 blocks of 256 threads (8 waves).
// ---------------------------------------------------------------------------
__global__ __launch_bounds__(256) void ori_wmma_kernel(const float* __restrict__ Q,
                                                       const float* __restrict__ K,
                                                       float* __restrict__ ORI) {
    const int wave = threadIdx.x >> 5;
    const int lane = threadIdx.x & 31;
    const int half = lane >> 4;
    const int lm   = lane & 15;

    const int tile = blockIdx.x * 8 + wave;   // 0 .. 1023
    const int b  = tile >> 8;                 // batch
    const int qt = (tile >> 4) & 15;          // q tile
    const int kt = tile & 15;                 // k tile
    const int q0 = qt << 4;
    const int k0 = kt << 4;

    const float* qbase = Q + (size_t)((b * NN + q0 + lm) * EE + 2 * half);
    const float* kbase = K + (size_t)((b * NN + k0 + lm) * EE + 2 * half);

    v8f c = {};
#pragma unroll
    for (int ec = 0; ec < 16; ++ec) {
        v2f a  = *(const v2f*)(qbase + ec * 4);
        v2f bv = *(const v2f*)(kbase + ec * 4);
        // (neg_a, A, neg_b, B, c_mod, C, reuse_a, reuse_b)
        c = __builtin_amdgcn_wmma_f32_16x16x4_f32(false, a, false, bv,
                                                  (short)0, c, false, false);
    }

    const float scale = 1.0f / 4096.0f;  // 1 / d_k^2
    float* obase = ORI + (size_t)b * (NN * NN) + (size_t)(q0 + half * 8) * NN + k0 + lm;
#pragma unroll
    for (int j = 0; j < 8; ++j) {
        obase[j * NN] = c[j] * scale;
    }
}

// ---------------------------------------------------------------------------
// Kernel 3 (bandwidth-dominant, ~777 MB streamed): one block per (b,r,q) row.
//   pla_score[k] = || h+r-t ||_2 over E=64
//   pla_w = softmax-ish(exp(-score)*mask1)           (row over k)
//   raa   = ori[b,q,k] * (1 + 0.5*pla_w)
//   out   = exp(raa)*mask2 / (rowsum + eps)
// Each 16-lane half-wave handles one k: 1 float4 load per tensor covers E=64.
// Wave reads 512 contiguous bytes per tensor per step -> fully coalesced.
// grid = B*R*N = 4096 blocks, 256 threads (8 waves).
// ---------------------------------------------------------------------------
__global__ __launch_bounds__(256) void fused_row_kernel(
    const float* __restrict__ h, const float* __restrict__ r,
    const float* __restrict__ t, const float* __restrict__ m1,
    const float* __restrict__ m2, const float* __restrict__ ori,
    float* __restrict__ out) {
    __shared__ float ek[NN];
    __shared__ float red[8];

    const int row    = blockIdx.x;           // (b*R + rr)*N + q
    const int b      = row >> 10;            // / (R*N)
    const int q      = row & (NN - 1);
    const size_t base_e = (size_t)row * (NN * EE);
    const int base_k    = row * NN;
    const int ori_base  = b * (NN * NN) + q * NN;

    const int lane = threadIdx.x & 31;
    const int wave = threadIdx.x >> 5;
    const int half = lane >> 4;
    const int lm   = lane & 15;

    // --- stage 1: exp(-||h+r-t||) * mask1 for all 256 k ---
    for (int s = 0; s < 16; ++s) {
        const int k = s * 16 + wave * 2 + half;
        const size_t off = base_e + (size_t)k * EE + lm * 4;
        const float4 h4 = *(const float4*)(h + off);
        const float4 r4 = *(const float4*)(r + off);
        const float4 t4 = *(const float4*)(t + off);
        const float dx = h4.x + r4.x - t4.x;
        const float dy = h4.y + r4.y - t4.y;
        const float dz = h4.z + r4.z - t4.z;
        const float dw = h4.w + r4.w - t4.w;
        float sq = dx * dx + dy * dy + dz * dz + dw * dw;
        sq += __shfl_xor(sq, 8, 16);
        sq += __shfl_xor(sq, 4, 16);
        sq += __shfl_xor(sq, 2, 16);
        sq += __shfl_xor(sq, 1, 16);
        if (lm == 0) {
            ek[k] = expf(-sqrtf(sq)) * m1[base_k + k];
        }
    }
    __syncthreads();

    // --- stage 2: row sum of ek over k ---
    float v = ek[threadIdx.x];
    v += __shfl_xor(v, 16, 32);
    v += __shfl_xor(v, 8, 32);
    v += __shfl_xor(v, 4, 32);
    v += __shfl_xor(v, 2, 32);
    v += __shfl_xor(v, 1, 32);
    if (lane == 0) red[wave] = v;
    __syncthreads();
    const float tot1 = red[0] + red[1] + red[2] + red[3] +
                       red[4] + red[5] + red[6] + red[7];

    // --- stage 3: raa scores + second normalization ---
    const int k = threadIdx.x;
    const float w   = ek[k] / (tot1 + EPS_C);
    const float raa = ori[ori_base + k] * (1.0f + LAMDA_C * w);
    const float e2  = expf(raa) * m2[base_k + k];

    __syncthreads();           // protect red[] before reuse
    float v2 = e2;
    v2 += __shfl_xor(v2, 16, 32);
    v2 += __shfl_xor(v2, 8, 32);
    v2 += __shfl_xor(v2, 4, 32);
    v2 += __shfl_xor(v2, 2, 32);
    v2 += __shfl_xor(v2, 1, 32);
    if (lane == 0) red[wave] = v2;
    __syncthreads();
    const float tot2 = red[0] + red[1] + red[2] + red[3] +
                       red[4] + red[5] + red[6] + red[7];

    out[base_k + k] = e2 / (tot2 + EPS_C);
}

// ---------------------------------------------------------------------------
// Launch
// Inputs: 0:n_emb 1:h_adj 2:r_adj 3:t_adj 4:mask_r_adj 5:mask_u_n_r_len 6:Wq 7:Wk
// Workspace layout (floats): Q[65536] | K[65536] | ORI[262144]  (~1.5 MB)
// ---------------------------------------------------------------------------
extern "C" void kernel_launch(void* const* d_in, const int* in_sizes, int n_in,
                              void* d_out, int out_size, void* d_ws, size_t ws_size,
                              hipStream_t stream) {
    const float* n_emb = (const float*)d_in[0];
    const float* h_adj = (const float*)d_in[1];
    const float* r_adj = (const float*)d_in[2];
    const float* t_adj = (const float*)d_in[3];
    const float* m1    = (const float*)d_in[4];
    const float* m2    = (const float*)d_in[5];
    const float* Wq    = (const float*)d_in[6];
    const float* Wk    = (const float*)d_in[7];
    float* out = (float*)d_out;

    float* Qw  = (float*)d_ws;                 // B*N*E = 65536 floats
    float* Kw  = Qw + BB * NN * EE;            // 65536 floats
    float* ORI = Kw + BB * NN * EE;            // B*N*N = 262144 floats

    qk_kernel<<<BB * NN, EE, 0, stream>>>(n_emb, Wq, Wk, Qw, Kw);
    ori_wmma_kernel<<<(BB * 16 * 16) / 8, 256, 0, stream>>>(Qw, Kw, ORI);
    fused_row_kernel<<<BB * RR * NN, 256, 0, stream>>>(h_adj, r_adj, t_adj,
                                                       m1, m2, ORI, out);
}